// BasicBlock_83897891160812
// MI455X (gfx1250) — compile-verified
//
#include <hip/hip_runtime.h>
#include <hip/hip_bf16.h>
#include <math.h>

#define NPTS   32768
#define NBP    8192
#define NBATCH 4
#define SDIM   128
#define HCAP   65536

#define XPITCH 66   // padded X-tile row stride (floats): A-frag loads ~conflict-free
#define WPITCH 80   // padded W-tile row stride (floats): B-frag loads conflict-free

typedef float v2f __attribute__((ext_vector_type(2)));
typedef float v8f __attribute__((ext_vector_type(8)));

__device__ __forceinline__ v8f wmma4(v2f a, v2f b, v8f c) {
  return __builtin_amdgcn_wmma_f32_16x16x4_f32(false, a, false, b, (short)0, c, false, false);
}

// ---------------------------------------------------------------------------
// WMMA GEMM, templated on NT = number of 16-column output tiles (branch-free
// inner loop). Y[N x Cout] = X[N x Cin] @ W[Cin x Cout] (+bias) (op1 = relu).
// 256 threads = 8 waves; each wave computes a 16-row x (NT*16)-col tile.
// ---------------------------------------------------------------------------
template <int NT>
__global__ void __launch_bounds__(256) k_gemm(const float* __restrict__ X, int ldx, int Cin,
                                              const float* __restrict__ W,
                                              const float* __restrict__ bias,
                                              float* __restrict__ Y, int ldy, int Cout, int op) {
  __shared__ float sW[64 * WPITCH];
  __shared__ float sX[8 * 16 * XPITCH];
  const int tid  = threadIdx.x;
  const int lane = tid & 31;
  const int wave = tid >> 5;
  float* sXw = &sX[wave * 16 * XPITCH];
  const int rowbase = blockIdx.x * 128 + wave * 16;
  const int m    = lane & 15;
  const int half = lane >> 4;
  v8f acc[NT];
#pragma unroll
  for (int n = 0; n < NT; n++) acc[n] = (v8f){0, 0, 0, 0, 0, 0, 0, 0};

  for (int kc = 0; kc < Cin; kc += 64) {
    __syncthreads();
    for (int idx = tid; idx < 64 * 64; idx += 256) {
      int r = idx >> 6, c = idx & 63;
      float wv = 0.0f;
      if ((kc + r) < Cin && c < Cout) wv = W[(size_t)(kc + r) * Cout + c];
      sW[r * WPITCH + c] = wv;
    }
    for (int rr = 0; rr < 16; rr++) {
      const float* xr = X + (size_t)(rowbase + rr) * ldx + kc;
      int c0 = lane, c1 = lane + 32;
      float v0 = ((kc + c0) < Cin) ? xr[c0] : 0.0f;
      float v1 = ((kc + c1) < Cin) ? xr[c1] : 0.0f;
      sXw[rr * XPITCH + c0] = v0;
      sXw[rr * XPITCH + c1] = v1;
    }
    __syncthreads();
    const int ktm = ((Cin - kc) >= 64) ? 16 : ((Cin - kc + 3) >> 2);
#pragma unroll 4
    for (int t = 0; t < ktm; t++) {
      int krow = 4 * t + 2 * half;
      v2f a;
      a.x = sXw[m * XPITCH + krow];
      a.y = sXw[m * XPITCH + krow + 1];
#pragma unroll
      for (int n = 0; n < NT; n++) {
        v2f b;
        b.x = sW[krow * WPITCH + n * 16 + m];
        b.y = sW[(krow + 1) * WPITCH + n * 16 + m];
        acc[n] = wmma4(a, b, acc[n]);
      }
    }
  }
  const int rb = rowbase + half * 8;
#pragma unroll
  for (int n = 0; n < NT; n++) {
    int col = n * 16 + m;
    if (col < Cout) {
      float bv = bias ? bias[col] : 0.0f;
#pragma unroll
      for (int r2 = 0; r2 < 8; r2++) {
        float v = acc[n][r2] + bv;
        if (op == 1) v = fmaxf(v, 0.0f);
        Y[(size_t)(rb + r2) * ldy + col] = v;
      }
    }
  }
}

// ---------------------------------------------------------------------------
// Submanifold 3x3x3 conv: Y = sum_k gather(F, nbr[:,k]) @ W27[k]  (raw, no BN)
// ---------------------------------------------------------------------------
__global__ void __launch_bounds__(256) k_conv(const float* __restrict__ F,
                                              const int* __restrict__ nbr,
                                              const float* __restrict__ W27,
                                              float* __restrict__ Y) {
  __shared__ float sW[64 * WPITCH];
  __shared__ float sX[8 * 16 * XPITCH];
  const int tid  = threadIdx.x;
  const int lane = tid & 31;
  const int wave = tid >> 5;
  float* sXw = &sX[wave * 16 * XPITCH];
  const int rowbase = blockIdx.x * 128 + wave * 16;
  const int m    = lane & 15;
  const int half = lane >> 4;
  v8f acc0 = {0,0,0,0,0,0,0,0}, acc1 = {0,0,0,0,0,0,0,0};
  v8f acc2 = {0,0,0,0,0,0,0,0}, acc3 = {0,0,0,0,0,0,0,0};

  for (int k = 0; k < 27; k++) {
    __syncthreads();
    for (int idx = tid; idx < 64 * 64; idx += 256) {
      int r = idx >> 6, c = idx & 63;
      sW[r * WPITCH + c] = W27[(size_t)k * 4096 + idx];
    }
    for (int rr = 0; rr < 16; rr++) {
      int nb = nbr[(size_t)(rowbase + rr) * 27 + k];   // broadcast load
      float v0 = 0.0f, v1 = 0.0f;
      if (nb >= 0) {
        const float* fr = F + (size_t)nb * 64;
        v0 = fr[lane];
        v1 = fr[lane + 32];
      }
      sXw[rr * XPITCH + lane] = v0;
      sXw[rr * XPITCH + lane + 32] = v1;
    }
    __syncthreads();
#pragma unroll 4
    for (int t = 0; t < 16; t++) {
      int krow = 4 * t + 2 * half;
      v2f a;
      a.x = sXw[m * XPITCH + krow];
      a.y = sXw[m * XPITCH + krow + 1];
      v2f b;
      b.x = sW[krow * WPITCH +  0 + m]; b.y = sW[(krow + 1) * WPITCH +  0 + m]; acc0 = wmma4(a, b, acc0);
      b.x = sW[krow * WPITCH + 16 + m]; b.y = sW[(krow + 1) * WPITCH + 16 + m]; acc1 = wmma4(a, b, acc1);
      b.x = sW[krow * WPITCH + 32 + m]; b.y = sW[(krow + 1) * WPITCH + 32 + m]; acc2 = wmma4(a, b, acc2);
      b.x = sW[krow * WPITCH + 48 + m]; b.y = sW[(krow + 1) * WPITCH + 48 + m]; acc3 = wmma4(a, b, acc3);
    }
  }
  const int rb = rowbase + half * 8;
#pragma unroll
  for (int r2 = 0; r2 < 8; r2++) {
    size_t ro = (size_t)(rb + r2) * 64;
    Y[ro +  0 + m] = acc0[r2];
    Y[ro + 16 + m] = acc1[r2];
    Y[ro + 32 + m] = acc2[r2];
    Y[ro + 48 + m] = acc3[r2];
  }
}

// ---------------------------------------------------------------------------
// BN column stats (sum, sumsq) over N rows.
// ---------------------------------------------------------------------------
__global__ void k_stats(const float* __restrict__ Y, int ld, int Cc, int Nrows,
                        float* __restrict__ stats) {
  int tid = threadIdx.x;
  int c = tid % Cc;
  int rsub = tid / Cc;
  int nr = blockDim.x / Cc;
  if (rsub >= nr) return;
  int chunk = (Nrows + gridDim.x - 1) / gridDim.x;
  int r0 = blockIdx.x * chunk;
  int r1 = r0 + chunk; if (r1 > Nrows) r1 = Nrows;
  float s = 0.0f, ss = 0.0f;
  for (int r = r0 + rsub; r < r1; r += nr) {
    float v = Y[(size_t)r * ld + c];
    s += v; ss += v * v;
  }
  atomicAdd(&stats[c], s);
  atomicAdd(&stats[Cc + c], ss);
}

// modes: 0=bn  1=bn,relu  2=relu(bn)+res  3=relu(bn+res)  4=relu(bn)*mul
__global__ void k_apply(const float* __restrict__ Y, int ld, int Cc, int Nrows,
                        const float* __restrict__ stats,
                        const float* __restrict__ g, const float* __restrict__ beta,
                        float* __restrict__ out, int ldo, int mode,
                        const float* __restrict__ res, int ldr,
                        const float* __restrict__ mul, int ldm) {
  size_t e = (size_t)blockIdx.x * blockDim.x + threadIdx.x;
  size_t tot = (size_t)Nrows * Cc;
  if (e >= tot) return;
  int r = (int)(e / Cc), c = (int)(e % Cc);
  float invn = 1.0f / (float)Nrows;
  float mean = stats[c] * invn;
  float var = stats[Cc + c] * invn - mean * mean;
  float v = (Y[(size_t)r * ld + c] - mean) * rsqrtf(var + 1e-5f) * g[c] + beta[c];
  if (mode == 1) v = fmaxf(v, 0.0f);
  else if (mode == 2) v = fmaxf(v, 0.0f) + res[(size_t)r * ldr + c];
  else if (mode == 3) v = fmaxf(v + res[(size_t)r * ldr + c], 0.0f);
  else if (mode == 4) v = fmaxf(v, 0.0f) * mul[(size_t)r * ldm + c];
  out[(size_t)r * ldo + c] = v;
}

// ---------------------------------------------------------------------------
// Small per-row kernels
// ---------------------------------------------------------------------------
__global__ void k_coordmin(const int* __restrict__ coords, int* __restrict__ minc) {
  int r = blockIdx.x * blockDim.x + threadIdx.x;
  if (r >= NPTS) return;
  atomicMin(&minc[0], coords[r * 3 + 0]);
  atomicMin(&minc[1], coords[r * 3 + 1]);
  atomicMin(&minc[2], coords[r * 3 + 2]);
}

__global__ void k_enh(const float* __restrict__ p9,
                      const float* __restrict__ caw1, const float* __restrict__ cab1,
                      const float* __restrict__ caw2, const float* __restrict__ cab2,
                      const float* __restrict__ naw1, const float* __restrict__ nab1,
                      const float* __restrict__ naw2, const float* __restrict__ nab2,
                      float* __restrict__ enh) {
  int r = blockIdx.x * blockDim.x + threadIdx.x;
  if (r >= NPTS) return;
  const float* p = &p9[(size_t)r * 9];
  float colf[3] = {p[3], p[4], p[5]};
  float nof[3]  = {p[6], p[7], p[8]};
  float h[16], ca[3], na[3];
#pragma unroll
  for (int j = 0; j < 16; j++) {
    float s = cab1[j];
    for (int i = 0; i < 3; i++) s += colf[i] * caw1[i * 16 + j];
    h[j] = fmaxf(s, 0.0f);
  }
#pragma unroll
  for (int k = 0; k < 3; k++) {
    float s = cab2[k];
    for (int j = 0; j < 16; j++) s += h[j] * caw2[j * 3 + k];
    ca[k] = 1.0f / (1.0f + expf(-s));
  }
#pragma unroll
  for (int j = 0; j < 16; j++) {
    float s = nab1[j];
    for (int i = 0; i < 3; i++) s += nof[i] * naw1[i * 16 + j];
    h[j] = fmaxf(s, 0.0f);
  }
#pragma unroll
  for (int k = 0; k < 3; k++) {
    float s = nab2[k];
    for (int j = 0; j < 16; j++) s += h[j] * naw2[j * 3 + k];
    na[k] = 1.0f / (1.0f + expf(-s));
  }
  float* o = &enh[(size_t)r * 9];
  o[0] = p[0]; o[1] = p[1]; o[2] = p[2];
  o[3] = colf[0] * ca[0]; o[4] = colf[1] * ca[1]; o[5] = colf[2] * ca[2];
  o[6] = nof[0] * na[0];  o[7] = nof[1] * na[1];  o[8] = nof[2] * na[2];
}

__global__ void k_safeat(const float* __restrict__ t32,
                         const float* __restrict__ w2, const float* __restrict__ b2,
                         const float* __restrict__ ff, const float* __restrict__ feat,
                         float* __restrict__ fu) {
  int r = blockIdx.x * blockDim.x + threadIdx.x;
  if (r >= NPTS) return;
  float s = b2[0];
  for (int j = 0; j < 32; j++) s += t32[(size_t)r * 32 + j] * w2[j];
  float sa = 1.0f / (1.0f + expf(-s));
  for (int c = 0; c < 64; c++) {
    size_t e = (size_t)r * 64 + c;
    fu[e] = ff[e] * sa + feat[e] * (1.0f - sa);
  }
}

__global__ void k_fjsmall(const float* __restrict__ X, const float* __restrict__ w2,
                          const float* __restrict__ b2, float* __restrict__ probs) {
  int r = blockIdx.x * blockDim.x + threadIdx.x;
  if (r >= NPTS) return;
  float l[3] = {b2[0], b2[1], b2[2]};
  for (int j = 0; j < 64; j++) {
    float f = X[(size_t)r * 64 + j];
    l[0] += f * w2[j * 3 + 0]; l[1] += f * w2[j * 3 + 1]; l[2] += f * w2[j * 3 + 2];
  }
  float mx = fmaxf(l[0], fmaxf(l[1], l[2]));
  float e0 = expf(l[0] - mx), e1 = expf(l[1] - mx), e2 = expf(l[2] - mx);
  float inv = 1.0f / (e0 + e1 + e2);
  probs[(size_t)r * 3 + 0] = e0 * inv;
  probs[(size_t)r * 3 + 1] = e1 * inv;
  probs[(size_t)r * 3 + 2] = e2 * inv;
}

__global__ void k_gs(const float* __restrict__ lin, const float* __restrict__ dens,
                     const float* __restrict__ probs, float* __restrict__ gsb,
                     float* __restrict__ gm) {
  int r = blockIdx.x * blockDim.x + threadIdx.x;
  if (r >= NPTS) return;
  float li = lin[r], de = dens[r];
  float p0 = probs[r * 3 + 0], p1 = probs[r * 3 + 1], p2 = probs[r * 3 + 2];
  float tower = (2.0f * de + p0) * (1.0f / 3.0f);
  float back  = (fmaxf(1.0f - li, 1.0f - de) + p1) * (1.0f / 3.0f);
  float line  = (2.0f * li + p2) * (1.0f / 3.0f);
  float g0 = tower * 4.0f + back * 16.0f + line * 2.0f  + 1e-6f;
  float g1 = tower * 4.0f + back * 16.0f + line * 2.0f  + 1e-6f;
  float g2 = tower * 4.0f + back * 16.0f + line * 10.0f + 1e-6f;
  gsb[r * 3 + 0] = g0; gsb[r * 3 + 1] = g1; gsb[r * 3 + 2] = g2;
  gm[r] = (g0 + g1 + g2) * (1.0f / 3.0f);
}

// stable rank of each element of gm; accumulate gs rows into rep buckets.
__global__ void __launch_bounds__(256) k_rank(const float* __restrict__ gm,
                                              const float* __restrict__ gsb,
                                              float* __restrict__ reps) {
  __shared__ float sg[256];
  int i = blockIdx.x * 256 + threadIdx.x;
  float vi = gm[i];
  int rank = 0;
  for (int t = 0; t < NPTS / 256; t++) {
    __syncthreads();
    sg[threadIdx.x] = gm[t * 256 + threadIdx.x];
    __syncthreads();
    int base = t * 256;
    for (int jj = 0; jj < 256; jj++) {
      float vj = sg[jj];
      int j = base + jj;
      rank += (vj < vi) || (vj == vi && j < i);
    }
  }
  float g0 = gsb[i * 3 + 0], g1 = gsb[i * 3 + 1], g2 = gsb[i * 3 + 2];
  if (rank >= 100 && rank < 200) {
    atomicAdd(&reps[0], g0); atomicAdd(&reps[1], g1); atomicAdd(&reps[2], g2);
  }
  if (rank >= NPTS - 100) {
    atomicAdd(&reps[3], g0); atomicAdd(&reps[4], g1); atomicAdd(&reps[5], g2);
  }
  if (rank < 100) {
    atomicAdd(&reps[6], g0); atomicAdd(&reps[7], g1); atomicAdd(&reps[8], g2);
  }
}

__global__ void k_repfin(float* reps) {
  int t = threadIdx.x;
  if (t < 9) reps[t] *= 0.01f;
}

// ---------------------------------------------------------------------------
// KNN (per batch): top-33 by (d2, idx), drop self, linearity + density
// ---------------------------------------------------------------------------
__global__ void __launch_bounds__(256) k_knn(const int* __restrict__ coords,
                                             float* __restrict__ lin,
                                             float* __restrict__ dens) {
  __shared__ float sx[256], sy[256], sz[256];
  const int b = blockIdx.y;
  const int il = blockIdx.x * 256 + threadIdx.x;
  const int gi = b * NBP + il;
  const float px = (float)coords[gi * 3 + 0];
  const float py = (float)coords[gi * 3 + 1];
  const float pz = (float)coords[gi * 3 + 2];
  float bd[33]; int bi[33];
  int cnt = 0, worst = 0;
  for (int t = 0; t < NBP / 256; t++) {
    __syncthreads();
    int gj = b * NBP + t * 256 + threadIdx.x;
    sx[threadIdx.x] = (float)coords[gj * 3 + 0];
    sy[threadIdx.x] = (float)coords[gj * 3 + 1];
    sz[threadIdx.x] = (float)coords[gj * 3 + 2];
    __syncthreads();
    int base = t * 256;
    for (int jj = 0; jj < 256; jj++) {
      float dx = sx[jj] - px, dy = sy[jj] - py, dz = sz[jj] - pz;
      float d2 = dx * dx + dy * dy + dz * dz;
      int j = base + jj;
      if (cnt < 33) {
        bd[cnt] = d2; bi[cnt] = j; cnt++;
        if (cnt == 33) {
          worst = 0;
          for (int s = 1; s < 33; s++)
            if (bd[s] > bd[worst] || (bd[s] == bd[worst] && bi[s] > bi[worst])) worst = s;
        }
      } else if (d2 < bd[worst] || (d2 == bd[worst] && j < bi[worst])) {
        bd[worst] = d2; bi[worst] = j;
        worst = 0;
        for (int s = 1; s < 33; s++)
          if (bd[s] > bd[worst] || (bd[s] == bd[worst] && bi[s] > bi[worst])) worst = s;
      }
    }
  }
  float nx[33], ny[33], nz[33];
  float sd = 0.0f, mx = 0.0f, my = 0.0f, mz = 0.0f;
  int nn = 0;
  for (int s = 0; s < 33; s++) {
    if (bi[s] == il) continue;  // self (d2 == 0, unique)
    int gj = b * NBP + bi[s];
    float qx = (float)coords[gj * 3 + 0];
    float qy = (float)coords[gj * 3 + 1];
    float qz = (float)coords[gj * 3 + 2];
    nx[nn] = qx; ny[nn] = qy; nz[nn] = qz; nn++;
    sd += sqrtf(bd[s]);
    mx += qx; my += qy; mz += qz;
  }
  const float invK = 1.0f / 32.0f;
  mx *= invK; my *= invK; mz *= invK;
  float cxx = 0, cxy = 0, cxz = 0, cyy = 0, cyz = 0, czz = 0;
  for (int s = 0; s < 32; s++) {
    float ax = nx[s] - mx, ay = ny[s] - my, az = nz[s] - mz;
    cxx += ax * ax; cxy += ax * ay; cxz += ax * az;
    cyy += ay * ay; cyz += ay * az; czz += az * az;
  }
  const float sc = 1.0f / 31.0f;
  cxx *= sc; cxy *= sc; cxz *= sc; cyy *= sc; cyz *= sc; czz *= sc;
  // eigenvalues of symmetric 3x3 (descending)
  float q = (cxx + cyy + czz) * (1.0f / 3.0f);
  float p1 = cxy * cxy + cxz * cxz + cyz * cyz;
  float aa = cxx - q, bb = cyy - q, cc = czz - q;
  float p2 = aa * aa + bb * bb + cc * cc + 2.0f * p1;
  float p = sqrtf(p2 * (1.0f / 6.0f));
  float e0, e1, e2;
  if (p < 1e-12f) {
    e0 = e1 = e2 = q;
  } else {
    float iv = 1.0f / p;
    float b00 = aa * iv, b11 = bb * iv, b22 = cc * iv;
    float b01 = cxy * iv, b02 = cxz * iv, b12 = cyz * iv;
    float detB = b00 * (b11 * b22 - b12 * b12)
               - b01 * (b01 * b22 - b12 * b02)
               + b02 * (b01 * b12 - b11 * b02);
    float rr = fminf(fmaxf(detB * 0.5f, -1.0f), 1.0f);
    float phi = acosf(rr) * (1.0f / 3.0f);
    e0 = q + 2.0f * p * cosf(phi);
    e2 = q + 2.0f * p * cosf(phi + 2.0943951023931953f);
    e1 = 3.0f * q - e0 - e2;
  }
  float ssum = e0 + e1 + e2;
  float isum = 1.0f / ssum;
  lin[gi]  = e0 * isum - e1 * isum - e2 * isum;
  dens[gi] = 1.0f / (sd * invK + 1e-6f);
}

// ---------------------------------------------------------------------------
// Clustering (group-by voxel row via hash table; slot index = segment id)
// ---------------------------------------------------------------------------
__global__ void k_cluster(const int* __restrict__ coords, const int* __restrict__ batch,
                          const int* __restrict__ minc, const float* __restrict__ rep,
                          unsigned long long* __restrict__ hk, int* __restrict__ cl) {
  int r = blockIdx.x * blockDim.x + threadIdx.x;
  if (r >= NPTS) return;
  float s0 = fmaxf(rep[0], 1e-6f), s1 = fmaxf(rep[1], 1e-6f), s2 = fmaxf(rep[2], 1e-6f);
  int vx = (int)floorf(((float)coords[r * 3 + 0] - (float)minc[0]) / s0);
  int vy = (int)floorf(((float)coords[r * 3 + 1] - (float)minc[1]) / s1);
  int vz = (int)floorf(((float)coords[r * 3 + 2] - (float)minc[2]) / s2);
  unsigned long long key =
      ((unsigned long long)(batch[r] + 1) << 59)
    ^ ((unsigned long long)(unsigned)vx * 0xD1B54A32D192ED03ull)
    ^ ((unsigned long long)(unsigned)vy * 0x94D049BB133111EBull)
    ^ ((unsigned long long)(unsigned)vz * 0x2545F4914F6CDD1Dull);
  if (key == ~0ull) key = 0x123456789ABCDEF1ull;
  unsigned slot = (unsigned)((key * 0x9E3779B97F4A7C15ull) >> 48) & (HCAP - 1);
  for (;;) {
    unsigned long long old = atomicCAS(&hk[slot], ~0ull, key);
    if (old == ~0ull || old == key) { cl[r] = (int)slot; return; }
    slot = (slot + 1) & (HCAP - 1);
  }
}

__global__ void k_segsum(const float* __restrict__ X, const int* __restrict__ cl,
                         float* __restrict__ seg, float* __restrict__ cnt, int addcnt) {
  size_t e = (size_t)blockIdx.x * blockDim.x + threadIdx.x;
  if (e >= (size_t)NPTS * 64) return;
  int r = (int)(e >> 6), c = (int)(e & 63);
  int s = cl[r];
  atomicAdd(&seg[(size_t)s * 64 + c], X[e]);
  if (addcnt && c == 0) atomicAdd(&cnt[s], 1.0f);
}

__global__ void k_segmeansub(float* __restrict__ X, const int* __restrict__ cl,
                             const float* __restrict__ seg, const float* __restrict__ cnt) {
  size_t e = (size_t)blockIdx.x * blockDim.x + threadIdx.x;
  if (e >= (size_t)NPTS * 64) return;
  int r = (int)(e >> 6), c = (int)(e & 63);
  int s = cl[r];
  X[e] -= seg[(size_t)s * 64 + c] / fmaxf(cnt[s], 1.0f);
}

__device__ __forceinline__ unsigned genc(float f) {
  unsigned u = __float_as_uint(f);
  return (u & 0x80000000u) ? ~u : (u | 0x80000000u);
}
__device__ __forceinline__ float gdec(unsigned k) {
  return (k & 0x80000000u) ? __uint_as_float(k & 0x7FFFFFFFu) : __uint_as_float(~k);
}

__global__ void k_gmax(const float* __restrict__ X, int n, unsigned* __restrict__ g) {
  int t = blockIdx.x * blockDim.x + threadIdx.x;
  unsigned best = 0u;
  for (int e = t; e < n; e += gridDim.x * blockDim.x) {
    unsigned k = genc(X[e]);
    if (k > best) best = k;
  }
  atomicMax(g, best);
}

__global__ void k_expseg(float* __restrict__ X, const unsigned* __restrict__ g,
                         const int* __restrict__ cl, float* __restrict__ seg) {
  size_t e = (size_t)blockIdx.x * blockDim.x + threadIdx.x;
  if (e >= (size_t)NPTS * 64) return;
  int r = (int)(e >> 6), c = (int)(e & 63);
  float mx = gdec(*g);
  float v = expf(X[e] - mx);
  X[e] = v;
  atomicAdd(&seg[(size_t)cl[r] * 64 + c], v);
}

__global__ void k_divseg(float* __restrict__ X, const int* __restrict__ cl,
                         const float* __restrict__ seg) {
  size_t e = (size_t)blockIdx.x * blockDim.x + threadIdx.x;
  if (e >= (size_t)NPTS * 64) return;
  int r = (int)(e >> 6), c = (int)(e & 63);
  X[e] = X[e] / (seg[(size_t)cl[r] * 64 + c] + 1e-6f);
}

__global__ void k_gatherseg(const int* __restrict__ cl, const float* __restrict__ seg,
                            float* __restrict__ F) {
  size_t e = (size_t)blockIdx.x * blockDim.x + threadIdx.x;
  if (e >= (size_t)NPTS * 64) return;
  int r = (int)(e >> 6), c = (int)(e & 63);
  F[e] = seg[(size_t)cl[r] * 64 + c];
}

__global__ void k_adpfuse(const float* __restrict__ fu, const float* __restrict__ adp,
                          const float* __restrict__ f0, const float* __restrict__ f1,
                          const float* __restrict__ f2, float* __restrict__ cat) {
  int r = blockIdx.x * blockDim.x + threadIdx.x;
  if (r >= NPTS) return;
  float l0 = 0.0f, l1 = 0.0f, l2 = 0.0f;
  for (int j = 0; j < 64; j++) {
    float f = fu[(size_t)r * 64 + j];
    l0 += f * adp[j * 3 + 0]; l1 += f * adp[j * 3 + 1]; l2 += f * adp[j * 3 + 2];
  }
  float mx = fmaxf(l0, fmaxf(l1, l2));
  float e0 = expf(l0 - mx), e1 = expf(l1 - mx), e2 = expf(l2 - mx);
  float inv = 1.0f / (e0 + e1 + e2);
  float a0 = e0 * inv, a1 = e1 * inv, a2 = e2 * inv;
  for (int c = 0; c < 64; c++) {
    size_t e = (size_t)r * 64 + c;
    cat[(size_t)r * 128 + 64 + c] = a0 * f0[e] + a1 * f1[e] + a2 * f2[e];
  }
}

// ---------------------------------------------------------------------------
// Voxel table + neighbor-index precompute for submanifold conv
// ---------------------------------------------------------------------------
__global__ void k_table(const int* __restrict__ coords, const int* __restrict__ batch,
                        int* __restrict__ tab) {
  int r = blockIdx.x * blockDim.x + threadIdx.x;
  if (r >= NPTS) return;
  int b = batch[r];
  int x = coords[r * 3 + 0], y = coords[r * 3 + 1], z = coords[r * 3 + 2];
  tab[(size_t)((b * SDIM + x) * SDIM + y) * SDIM + z] = r;
}

__global__ void k_nbr(const int* __restrict__ coords, const int* __restrict__ batch,
                      const int* __restrict__ tab, int* __restrict__ nbr) {
  int t = blockIdx.x * blockDim.x + threadIdx.x;
  if (t >= NPTS * 27) return;
  int i = t / 27, k = t - i * 27;
  int dx = k / 9 - 1;
  int r9 = k % 9;
  int dy = r9 / 3 - 1;
  int dz = r9 % 3 - 1;
  int x = coords[i * 3 + 0] + dx;
  int y = coords[i * 3 + 1] + dy;
  int z = coords[i * 3 + 2] + dz;
  int out = -1;
  if (x >= 0 && x < SDIM && y >= 0 && y < SDIM && z >= 0 && z < SDIM) {
    int b = batch[i];
    out = tab[(size_t)((b * SDIM + x) * SDIM + y) * SDIM + z];
  }
  nbr[t] = out;
}

// ---------------------------------------------------------------------------
// Host-side orchestration
// ---------------------------------------------------------------------------
extern "C" void kernel_launch(void* const* d_in, const int* in_sizes, int n_in,
                              void* d_out, int out_size, void* d_ws, size_t ws_size,
                              hipStream_t stream) {
  (void)in_sizes; (void)n_in; (void)out_size; (void)ws_size;

  const float* feat      = (const float*)d_in[0];
  const int*   coords    = (const int*)  d_in[1];
  const int*   batch     = (const int*)  d_in[2];
  const float* cm_fp_w   = (const float*)d_in[3];
  const float* cm_fp_b   = (const float*)d_in[4];
  const float* cm_fp_g   = (const float*)d_in[5];
  const float* cm_fp_be  = (const float*)d_in[6];
  const float* cm_ca_w1  = (const float*)d_in[7];
  const float* cm_ca_b1  = (const float*)d_in[8];
  const float* cm_ca_w2  = (const float*)d_in[9];
  const float* cm_ca_b2  = (const float*)d_in[10];
  const float* cm_na_w1  = (const float*)d_in[11];
  const float* cm_na_b1  = (const float*)d_in[12];
  const float* cm_na_w2  = (const float*)d_in[13];
  const float* cm_na_b2  = (const float*)d_in[14];
  const float* cm_ff_w1  = (const float*)d_in[15];
  const float* cm_ff_b1  = (const float*)d_in[16];
  const float* cm_ff_g   = (const float*)d_in[17];
  const float* cm_ff_be  = (const float*)d_in[18];
  const float* cm_ff_w2  = (const float*)d_in[19];
  const float* cm_ff_b2  = (const float*)d_in[20];
  const float* cm_sa_w1  = (const float*)d_in[21];
  const float* cm_sa_b1  = (const float*)d_in[22];
  const float* cm_sa_w2  = (const float*)d_in[23];
  const float* cm_sa_b2  = (const float*)d_in[24];
  const float* fj_w1     = (const float*)d_in[25];
  const float* fj_b1     = (const float*)d_in[26];
  const float* fj_g      = (const float*)d_in[27];
  const float* fj_be     = (const float*)d_in[28];
  const float* fj_w2     = (const float*)d_in[29];
  const float* fj_b2     = (const float*)d_in[30];
  const float* proj_w    = (const float*)d_in[31];
  const float* proj_g    = (const float*)d_in[32];
  const float* proj_be   = (const float*)d_in[33];
  const float* lw_w      = (const float*)d_in[34];
  const float* lw_g      = (const float*)d_in[35];
  const float* lw_be     = (const float*)d_in[36];
  const float* wt_w      = (const float*)d_in[37];
  const float* adp_w     = (const float*)d_in[38];
  const float* fuse_w    = (const float*)d_in[39];
  const float* fuse_g    = (const float*)d_in[40];
  const float* fuse_be   = (const float*)d_in[41];
  const float* conv1_w   = (const float*)d_in[42];
  const float* bn1_g     = (const float*)d_in[43];
  const float* bn1_b     = (const float*)d_in[44];
  const float* conv2_w   = (const float*)d_in[45];
  const float* bn2_g     = (const float*)d_in[46];
  const float* bn2_b     = (const float*)d_in[47];
  float* outp = (float*)d_out;

  // workspace layout
  char* cur = (char*)d_ws;
  auto alloc = [&](size_t sz) -> void* {
    void* p = (void*)cur;
    cur += (sz + 255) & ~(size_t)255;
    return p;
  };
  const size_t BIG = (size_t)NPTS * 64 * sizeof(float);
  float* A   = (float*)alloc(BIG);
  float* Bb  = (float*)alloc(BIG);
  float* PW  = (float*)alloc(BIG);
  float* FF  = (float*)alloc(BIG);
  float* FU  = (float*)alloc(BIG);
  float* RES = (float*)alloc(BIG);
  float* V1  = (float*)alloc(BIG);
  float* F0  = (float*)alloc(BIG);
  float* F1  = (float*)alloc(BIG);
  float* F2  = (float*)alloc(BIG);
  float* CAT = (float*)alloc((size_t)NPTS * 128 * sizeof(float));
  float* SEG = (float*)alloc((size_t)HCAP * 64 * sizeof(float));
  float* SEGC = (float*)alloc((size_t)HCAP * sizeof(float));
  unsigned long long* HK = (unsigned long long*)alloc((size_t)HCAP * 8);
  int*   TAB = (int*)alloc((size_t)NBATCH * SDIM * SDIM * SDIM * sizeof(int));
  int*   NBR = (int*)alloc((size_t)NPTS * 27 * sizeof(int));
  float* P9R = (float*)alloc((size_t)NPTS * 9 * sizeof(float));
  float* P9A = (float*)alloc((size_t)NPTS * 9 * sizeof(float));
  float* ENH = (float*)alloc((size_t)NPTS * 9 * sizeof(float));
  float* T32 = (float*)alloc((size_t)NPTS * 32 * sizeof(float));
  float* PROBS = (float*)alloc((size_t)NPTS * 3 * sizeof(float));
  float* GSB = (float*)alloc((size_t)NPTS * 3 * sizeof(float));
  float* LIN = (float*)alloc((size_t)NPTS * sizeof(float));
  float* DEN = (float*)alloc((size_t)NPTS * sizeof(float));
  float* GM  = (float*)alloc((size_t)NPTS * sizeof(float));
  int*   CL  = (int*)alloc((size_t)NPTS * sizeof(int));
  float* STATS = (float*)alloc(512);
  float* REPS  = (float*)alloc(64);
  int*   MINC  = (int*)alloc(16);
  unsigned* GMX = (unsigned*)alloc(16);

  const int EB = (NPTS * 64) / 256;   // elementwise blocks (N x 64)
  const int RB = NPTS / 256;          // per-row blocks

  auto gemm = [&](const float* X, int ldx, int Cin, const float* W, const float* bias,
                  float* Y, int ldy, int Cout, int op) {
    if (Cout <= 16)
      k_gemm<1><<<dim3(NPTS / 128), dim3(256), 0, stream>>>(X, ldx, Cin, W, bias, Y, ldy, Cout, op);
    else if (Cout <= 32)
      k_gemm<2><<<dim3(NPTS / 128), dim3(256), 0, stream>>>(X, ldx, Cin, W, bias, Y, ldy, Cout, op);
    else
      k_gemm<4><<<dim3(NPTS / 128), dim3(256), 0, stream>>>(X, ldx, Cin, W, bias, Y, ldy, Cout, op);
  };
  auto bnpass = [&](const float* Y, int ld, int Cc, const float* g, const float* be,
                    float* out, int ldo, int mode, const float* res, int ldr,
                    const float* mul, int ldm) {
    (void)hipMemsetAsync(STATS, 0, 2 * 64 * sizeof(float), stream);
    k_stats<<<dim3(128), dim3(256), 0, stream>>>(Y, ld, Cc, NPTS, STATS);
    int tot = NPTS * Cc;
    k_apply<<<dim3((tot + 255) / 256), dim3(256), 0, stream>>>(
        Y, ld, Cc, NPTS, STATS, g, be, out, ldo, mode, res, ldr, mul, ldm);
  };

  // --- coord min (for clustering) ---
  (void)hipMemsetAsync(MINC, 0x7F, 12, stream);
  k_coordmin<<<dim3(RB), dim3(256), 0, stream>>>(coords, MINC);

  // --- p = relu(bn(feat @ cm_fp_w + b)) ---
  gemm(feat, 64, 64, cm_fp_w, cm_fp_b, P9R, 9, 9, 0);
  bnpass(P9R, 9, 9, cm_fp_g, cm_fp_be, P9A, 9, 1, nullptr, 0, nullptr, 0);
  k_enh<<<dim3(RB), dim3(256), 0, stream>>>(P9A, cm_ca_w1, cm_ca_b1, cm_ca_w2, cm_ca_b2,
                                            cm_na_w1, cm_na_b1, cm_na_w2, cm_na_b2, ENH);

  // --- ff = relu(bn(enh @ w1 + b1)) @ w2 + b2 ---
  gemm(ENH, 9, 9, cm_ff_w1, cm_ff_b1, A, 64, 64, 0);
  bnpass(A, 64, 64, cm_ff_g, cm_ff_be, Bb, 64, 1, nullptr, 0, nullptr, 0);
  gemm(Bb, 64, 64, cm_ff_w2, cm_ff_b2, FF, 64, 64, 0);

  // --- sa gate; feat update ---
  gemm(FF, 64, 64, cm_sa_w1, cm_sa_b1, T32, 32, 32, 1);
  k_safeat<<<dim3(RB), dim3(256), 0, stream>>>(T32, cm_sa_w2, cm_sa_b2, FF, feat, FU);

  // --- KNN geometry ---
  k_knn<<<dim3(NBP / 256, NBATCH), dim3(256), 0, stream>>>(coords, LIN, DEN);

  // --- fj logits -> probs ---
  gemm(FU, 64, 64, fj_w1, fj_b1, A, 64, 64, 0);
  bnpass(A, 64, 64, fj_g, fj_be, Bb, 64, 1, nullptr, 0, nullptr, 0);
  k_fjsmall<<<dim3(RB), dim3(256), 0, stream>>>(Bb, fj_w2, fj_b2, PROBS);
  k_gs<<<dim3(RB), dim3(256), 0, stream>>>(LIN, DEN, PROBS, GSB, GM);

  // --- reps via stable rank ---
  (void)hipMemsetAsync(REPS, 0, 9 * sizeof(float), stream);
  k_rank<<<dim3(RB), dim3(256), 0, stream>>>(GM, GSB, REPS);
  k_repfin<<<dim3(1), dim3(32), 0, stream>>>(REPS);

  // --- three pooling branches ---
  float* FEATS[3] = {F0, F1, F2};
  for (int i = 0; i < 3; i++) {
    (void)hipMemsetAsync(HK, 0xFF, (size_t)HCAP * 8, stream);
    k_cluster<<<dim3(RB), dim3(256), 0, stream>>>(coords, batch, MINC, REPS + 3 * i, HK, CL);

    gemm(FU, 64, 64, lw_w + (size_t)i * 4096, nullptr, A, 64, 64, 0);
    bnpass(A, 64, 64, lw_g + i * 64, lw_be + i * 64, PW, 64, 1, nullptr, 0, nullptr, 0);

    (void)hipMemsetAsync(SEG, 0, (size_t)HCAP * 64 * sizeof(float), stream);
    (void)hipMemsetAsync(SEGC, 0, (size_t)HCAP * sizeof(float), stream);
    k_segsum<<<dim3(EB), dim3(256), 0, stream>>>(PW, CL, SEG, SEGC, 1);
    k_segmeansub<<<dim3(EB), dim3(256), 0, stream>>>(PW, CL, SEG, SEGC);

    gemm(PW, 64, 64, wt_w + (size_t)i * 4096, nullptr, Bb, 64, 64, 0);

    (void)hipMemsetAsync(GMX, 0, sizeof(unsigned), stream);
    k_gmax<<<dim3(256), dim3(256), 0, stream>>>(Bb, NPTS * 64, GMX);
    (void)hipMemsetAsync(SEG, 0, (size_t)HCAP * 64 * sizeof(float), stream);
    k_expseg<<<dim3(EB), dim3(256), 0, stream>>>(Bb, GMX, CL, SEG);
    k_divseg<<<dim3(EB), dim3(256), 0, stream>>>(Bb, CL, SEG);

    gemm(FU, 64, 64, proj_w + (size_t)i * 4096, nullptr, A, 64, 64, 0);
    bnpass(A, 64, 64, proj_g + i * 64, proj_be + i * 64, PW, 64, 4, nullptr, 0, Bb, 64);

    (void)hipMemsetAsync(SEG, 0, (size_t)HCAP * 64 * sizeof(float), stream);
    k_segsum<<<dim3(EB), dim3(256), 0, stream>>>(PW, CL, SEG, SEGC, 0);
    k_gatherseg<<<dim3(EB), dim3(256), 0, stream>>>(CL, SEG, FEATS[i]);
  }

  // --- adaptive fusion into cat[:,64:128] ---
  k_adpfuse<<<dim3(RB), dim3(256), 0, stream>>>(FU, adp_w, F0, F1, F2, CAT);
  // --- fl into cat[:,0:64] ---
  gemm(FU, 64, 64, proj_w + (size_t)3 * 4096, nullptr, A, 64, 64, 0);
  bnpass(A, 64, 64, proj_g + 192, proj_be + 192, CAT, 128, 1, nullptr, 0, nullptr, 0);
  // --- h = relu(bn(cat @ fuse_w)) + feat ---
  gemm(CAT, 128, 128, fuse_w, nullptr, A, 64, 64, 0);
  bnpass(A, 64, 64, fuse_g, fuse_be, RES, 64, 2, FU, 64, nullptr, 0);

  // --- voxel table + neighbor indices ---
  (void)hipMemsetAsync(TAB, 0xFF, (size_t)NBATCH * SDIM * SDIM * SDIM * sizeof(int), stream);
  k_table<<<dim3(RB), dim3(256), 0, stream>>>(coords, batch, TAB);
  k_nbr<<<dim3((NPTS * 27 + 255) / 256), dim3(256), 0, stream>>>(coords, batch, TAB, NBR);

  // --- conv1 + bn + relu ---
  k_conv<<<dim3(NPTS / 128), dim3(256), 0, stream>>>(RES, NBR, conv1_w, A);
  bnpass(A, 64, 64, bn1_g, bn1_b, V1, 64, 1, nullptr, 0, nullptr, 0);
  // --- conv2 + bn, + res, relu -> output ---
  k_conv<<<dim3(NPTS / 128), dim3(256), 0, stream>>>(V1, NBR, conv2_w, A);
  bnpass(A, 64, 64, bn2_g, bn2_b, outp, 64, 3, RES, 64, nullptr, 0);
}